// MIPMap_20306605375622
// MI455X (gfx1250) — compile-verified
//
#include <hip/hip_runtime.h>

// ---------------------------------------------------------------------------
// MIPMap lookup, width=0.01  ->  blend of bilinear taps on mip L3 (128^2) and
// L4 (64^2) with delta = 10 + log2(0.01) - 3 = 0.3561438102252756.
//
// Bandwidth-bound: ~84 MB mandatory HBM traffic -> ~3.6us floor at 23.3 TB/s.
// The ~400 MB of random texel gathers are served from LDS: both mip levels
// (240 KB) are staged per-block into CDNA5's 320 KB WGP LDS via the async
// global->LDS data mover (global_load_async_to_lds_b128 + s_wait_asynccnt),
// then all taps are ds_load gathers at LDS bandwidth.
// ---------------------------------------------------------------------------

#define L1H 512
#define L2H 256
#define L3H 128
#define L4H 64

// workspace layout (in floats)
#define L1_OFF 0
#define L2_OFF (L1H * L1H * 3)            // 786432
#define L3_OFF (L2_OFF + L2H * L2H * 3)   // 983040  (byte offset % 16 == 0)
#define L4_OFF (L3_OFF + L3H * L3H * 3)   // 1032192 (contiguous after L3)

#define L3_FLOATS (L3H * L3H * 3)                 // 49152
#define L4_FLOATS (L4H * L4H * 3)                 // 12288
#define STAGE_FLOATS (L3_FLOATS + L4_FLOATS)      // 61440
#define STAGE_BYTES (STAGE_FLOATS * 4)            // 245760 < 320KB WGP LDS

#define NPTS (2048 * 2048)
#define NPAIRS (NPTS / 2)
#define DELTA 0.3561438102252756f

#define LOOKUP_BLOCKS 512
#define LOOKUP_THREADS 1024

// Use gfx1250 async global->LDS copies when the toolchain exposes them.
#if defined(__HIP_DEVICE_COMPILE__) &&                                         \
    __has_builtin(__builtin_amdgcn_global_load_async_to_lds_b128) &&           \
    __has_builtin(__builtin_amdgcn_s_wait_asynccnt)
#define USE_ASYNC_LDS 1
#else
#define USE_ASYNC_LDS 0
#endif

#if USE_ASYNC_LDS
typedef int v4i_t __attribute__((ext_vector_type(4)));
typedef __attribute__((address_space(1))) v4i_t g_v4i;  // global-memory int4
typedef __attribute__((address_space(3))) v4i_t l_v4i;  // LDS int4
#endif

// ------------------------- bilinear tap (triangle) -------------------------
// Faithful to the reference: weight (1-ds)*dt pairs with texel (s0+1, t0) and
// ds*(1-dt) with (s0, t0+1); wrap idx==n -> 0, idx==-1 -> n-1.
__device__ __forceinline__ void bilerp3(const float* __restrict__ tex, int n,
                                        float sc, float tc, float r[3]) {
  float s = sc * (float)n - 0.5f;
  float t = tc * (float)n - 0.5f;
  float sf = floorf(s), tf = floorf(t);
  int s0 = (int)sf, t0 = (int)tf;
  float ds = s - sf, dt = t - tf;
  int s0w = (s0 < 0) ? s0 + n : s0;
  int s1w = (s0 + 1 == n) ? 0 : s0 + 1;
  int t0w = (t0 < 0) ? t0 + n : t0;
  int t1w = (t0 + 1 == n) ? 0 : t0 + 1;
  const float* p00 = tex + (s0w * n + t0w) * 3;
  const float* p10 = tex + (s1w * n + t0w) * 3;
  const float* p01 = tex + (s0w * n + t1w) * 3;
  const float* p11 = tex + (s1w * n + t1w) * 3;
  float w00 = (1.0f - ds) * (1.0f - dt);
  float w10 = (1.0f - ds) * dt;   // reference pairs this with (s0+1, t0)
  float w01 = ds * (1.0f - dt);   // reference pairs this with (s0, t0+1)
  float w11 = ds * dt;
#pragma unroll
  for (int c = 0; c < 3; ++c)
    r[c] = w00 * p00[c] + w10 * p10[c] + w01 * p01[c] + w11 * p11[c];
}

__device__ __forceinline__ void lookup_one(const float* __restrict__ s3,
                                           const float* __restrict__ s4,
                                           float sc, float tc,
                                           float* __restrict__ o) {
  float lo[3], hi[3];
  bilerp3(s3, L3H, sc, tc, lo);
  bilerp3(s4, L4H, sc, tc, hi);
  o[0] = lo[0] + DELTA * (hi[0] - lo[0]);
  o[1] = lo[1] + DELTA * (hi[1] - lo[1]);
  o[2] = lo[2] + DELTA * (hi[2] - lo[2]);
}

// ------------------------------ main lookup --------------------------------
// tables = L3 followed immediately by L4 in global memory (contiguous).
// First in the file so the disasm snippet shows the async staging loop.
__global__ __launch_bounds__(LOOKUP_THREADS) void mip_lookup(
    const float* __restrict__ st, const float* __restrict__ tables,
    float* __restrict__ out) {
  extern __shared__ float smem[];  // STAGE_BYTES dynamic LDS
  const int tid = threadIdx.x;

  // ---- stage L3 + L4 (245,760 B) into LDS ----
#if USE_ASYNC_LDS
  {
    const char* gsrc = (const char*)tables;
    char* ldst = (char*)smem;
    for (int off = tid * 16; off < STAGE_BYTES; off += LOOKUP_THREADS * 16) {
      __builtin_amdgcn_global_load_async_to_lds_b128(
          (g_v4i*)(gsrc + off), (l_v4i*)(ldst + off),
          /*offset=*/0, /*cpol=*/0);
    }
    __builtin_amdgcn_s_wait_asynccnt(0);
  }
#else
  {
    const float4* gsrc = (const float4*)tables;
    float4* ldst = (float4*)smem;
    for (int i = tid; i < STAGE_FLOATS / 4; i += LOOKUP_THREADS) {
      ldst[i] = gsrc[i];
    }
  }
#endif
  __syncthreads();

  const float* s3 = smem;
  const float* s4 = smem + L3_FLOATS;

  // Two points per thread per iteration: one coalesced b128 read of st,
  // 24 contiguous output bytes per thread.
  const float4* st4 = (const float4*)st;
  const unsigned stride = gridDim.x * (unsigned)LOOKUP_THREADS;
  for (unsigned p = blockIdx.x * (unsigned)LOOKUP_THREADS + tid; p < NPAIRS;
       p += stride) {
    float4 q = st4[p];  // (s0,t0,s1,t1) for points 2p, 2p+1
    float o[6];
    lookup_one(s3, s4, q.x, q.y, o);
    lookup_one(s3, s4, q.z, q.w, o + 3);
    float* dst = out + 6u * p;
#pragma unroll
    for (int k = 0; k < 6; ++k) dst[k] = o[k];
  }
}

// --------------------------- 2x2 mean pooling ------------------------------
// Compile-time dims: div/mod lower to shifts; one thread per output texel.
template <int DH, int DW>
__global__ void downsample2x2(const float* __restrict__ src,
                              float* __restrict__ dst) {
  int i = blockIdx.x * blockDim.x + threadIdx.x;  // output texel index
  if (i >= DH * DW) return;
  int x = i % DW;
  int y = i / DW;
  const int sw3 = DW * 2 * 3;  // source row stride in floats
  const float* s = src + (2 * y) * sw3 + (2 * x) * 3;
  float* d = dst + i * 3;
#pragma unroll
  for (int c = 0; c < 3; ++c)
    d[c] = 0.25f * ((s[c] + s[3 + c]) + (s[sw3 + c] + s[sw3 + 3 + c]));
}

// ------------------------------- launcher ----------------------------------
extern "C" void kernel_launch(void* const* d_in, const int* in_sizes, int n_in,
                              void* d_out, int out_size, void* d_ws,
                              size_t ws_size, hipStream_t stream) {
  const float* st = (const float*)d_in[0];   // [2048,2048,2] f32
  const float* img = (const float*)d_in[1];  // [1024,1024,3] f32
  float* ws = (float*)d_ws;
  float* L1 = ws + L1_OFF;
  float* L2 = ws + L2_OFF;
  float* L3 = ws + L3_OFF;
  float* L4 = ws + L4_OFF;  // contiguous with L3

  const int T = 256;
  downsample2x2<L1H, L1H><<<(L1H * L1H + T - 1) / T, T, 0, stream>>>(img, L1);
  downsample2x2<L2H, L2H><<<(L2H * L2H + T - 1) / T, T, 0, stream>>>(L1, L2);
  downsample2x2<L3H, L3H><<<(L3H * L3H + T - 1) / T, T, 0, stream>>>(L2, L3);
  downsample2x2<L4H, L4H><<<(L4H * L4H + T - 1) / T, T, 0, stream>>>(L3, L4);

  // 240KB dynamic LDS per block (one block per WGP); raise the limit.
  (void)hipFuncSetAttribute((const void*)mip_lookup,
                            hipFuncAttributeMaxDynamicSharedMemorySize,
                            STAGE_BYTES);
  mip_lookup<<<LOOKUP_BLOCKS, LOOKUP_THREADS, STAGE_BYTES, stream>>>(
      st, L3, (float*)d_out);
}